// USMSharp_15771119911620
// MI455X (gfx1250) — compile-verified
//
#include <hip/hip_runtime.h>
#include <hip/hip_bf16.h>

// USM sharpen for (16,512,512,3) f32, 51x51 separable Gaussian (radius 25).
// Convolution mapped onto V_WMMA_F32_16X16X4_F32 as Toeplitz GEMM:
//   16x16 output tile = sum over 17 K=4 steps of A(16x4) x B(4x16).
// Toeplitz weight fragments are tile-invariant -> precomputed per block into
// an LDS table so the inner loop is ds_load_b64 + ds_load_b64 + wmma.

typedef float v2f __attribute__((ext_vector_type(2)));
typedef float v8f __attribute__((ext_vector_type(8)));

#define IMG_W   512
#define PLANE   (IMG_W * IMG_W)      // 262144 floats per channel plane
#define NPLANES 48                   // 16 batches * 3 channels
#define KSIZE   51
#define RAD     25
#define TW      68                   // staged tile width: 66 needed, pad to 17*4
#define NSTEP   17                   // ceil(66/4) K-steps of 4
#define WAVES_PER_BLOCK 8
#define TILES_TOTAL (NPLANES * 32 * 32)   // 49152 16x16 tiles per pass

__device__ __forceinline__ int reflect511(int x) {
    // jnp.pad 'reflect': -1 -> 1, 512 -> 510   (|x| <= ~43 past edge here)
    if (x < 0) x = -x;
    if (x > IMG_W - 1) x = 2 * (IMG_W - 1) - x;
    return x;
}

// Fill the tile-invariant Toeplitz weight fragment table:
//   wfrag[s][lane] = ( g[4s+kk-n], g[4s+kk-n+1] ),  n = lane&15, kk = 2*(lane>>4)
// Used as the B fragment (h-pass) and the A fragment (v-pass) — same values.
__device__ __forceinline__ void fill_wfrag(v2f (*wfrag)[32], const float* __restrict__ g1,
                                           int tid) {
    if (tid < 32) {
        const int n  = tid & 15;
        const int kk = (tid >> 4) * 2;
        #pragma unroll
        for (int s = 0; s < NSTEP; ++s) {
            const int t0 = 4 * s + kk - n;
            v2f w;
            w.x = (t0 >= 0     && t0     < KSIZE) ? g1[t0]     : 0.f;
            w.y = (t0 + 1 >= 0 && t0 + 1 < KSIZE) ? g1[t0 + 1] : 0.f;
            wfrag[s][tid] = w;
        }
    }
}

// ---------------------------------------------------------------------------
// 1) Recover 1-D Gaussian from the 2-D outer product: g[i] = sum_j k2d[i][j]
//    (exact because sum(g)==1).
__global__ void usm_gauss1d(const float* __restrict__ k2d, float* __restrict__ g1) {
    int i = threadIdx.x;
    if (i < KSIZE) {
        float s = 0.f;
        for (int j = 0; j < KSIZE; ++j) s += k2d[i * KSIZE + j];
        g1[i] = s;
    } else if (i < 64) {
        g1[i] = 0.f;
    }
}

// ---------------------------------------------------------------------------
// 2) NHWC -> planar (plane = b*3 + c). Reads coalesced; writes hit L2.
__global__ __launch_bounds__(256) void usm_nhwc2planar(const float* __restrict__ img,
                                                       float* __restrict__ P) {
    int i = blockIdx.x * 256 + threadIdx.x;       // 0 .. 16*512*512*3-1
    int c = i % 3;
    int q = i / 3;
    int x = q & (IMG_W - 1);  q >>= 9;
    int y = q & (IMG_W - 1);
    int b = q >> 9;
    P[(b * 3 + c) * PLANE + y * IMG_W + x] = img[i];
}

// ---------------------------------------------------------------------------
// 3) Horizontal 51-tap blur via WMMA.
//    Wave computes one 16x16 tile; 16x68 input patch staged in LDS (reflect-x).
//    A[m][k] = patch[m][base+k]  (image),  B[k][n] = wfrag (Toeplitz weights).
__global__ __launch_bounds__(256) void usm_hblur(const float* __restrict__ in,
                                                 float* __restrict__ outp,
                                                 const float* __restrict__ g1) {
    __shared__ __align__(16) v2f   wfrag[NSTEP][32];
    __shared__ __align__(16) float patch[WAVES_PER_BLOCK][16 * TW];

    const int tid  = threadIdx.x;
    const int wave = tid >> 5;
    const int lane = tid & 31;
    fill_wfrag(wfrag, g1, tid);

    const int gw    = blockIdx.x * WAVES_PER_BLOCK + wave;  // tile id
    const int plane = gw >> 10;
    const int ty    = (gw >> 5) & 31;
    const int tx    = gw & 31;
    const int y0    = ty * 16, x0 = tx * 16;

    const float* pl = in + plane * PLANE;
    float* pw = patch[wave];
    // Stage 16 rows x 68 cols. Cols 66/67 get valid (reflected) data so the
    // weight-0 WMMA terms multiply finite values, never garbage.
    if (tx >= 2 && tx <= 29) {                 // interior: no reflection possible
        const float* src = pl + y0 * IMG_W + (x0 - RAD);
        for (int i = lane; i < 16 * TW; i += 32) {
            int rr = i / TW, cc = i - rr * TW;
            pw[i] = src[rr * IMG_W + cc];
        }
    } else {
        for (int i = lane; i < 16 * TW; i += 32) {
            int rr = i / TW, cc = i - rr * TW;
            int x = reflect511(x0 - RAD + cc);
            pw[i] = pl[(y0 + rr) * IMG_W + x];
        }
    }
    __syncthreads();

    const int m  = lane & 15;
    const int hi = lane >> 4;
    const int kk = hi * 2;           // this half-wave holds K = kk, kk+1

    v8f acc = {0.f, 0.f, 0.f, 0.f, 0.f, 0.f, 0.f, 0.f};
    const float* arow = &pw[m * TW + kk];
    #pragma unroll
    for (int s = 0; s < NSTEP; ++s) {
        v2f a = *(const v2f*)(arow + 4 * s);               // ds_load_b64
        v2f b = wfrag[s][lane];                            // ds_load_b64
        acc = __builtin_amdgcn_wmma_f32_16x16x4_f32(
            false, a, false, b, (short)0, acc, false, false);
    }

    float* po = outp + plane * PLANE;
    #pragma unroll
    for (int r = 0; r < 8; ++r)                            // D: VGPR r -> M=r / r+8
        po[(y0 + r + 8 * hi) * IMG_W + x0 + m] = acc[r];
}

// ---------------------------------------------------------------------------
// 4/6) Vertical 51-tap blur via WMMA (weights in A, image rows in B), fused
//      with the USM epilogue.
//  MODE 0: blur of img -> residual to Rbuf, threshold mask to outBuf (planar).
//  MODE 1: blur of mask = soft_mask -> final blend, scattered to NHWC outBuf.
template <int MODE>
__global__ __launch_bounds__(256) void usm_vblur(const float* __restrict__ in,
                                                 const float* __restrict__ Pimg,
                                                 const float* __restrict__ g1,
                                                 float* __restrict__ Rbuf,
                                                 float* __restrict__ outBuf) {
    __shared__ __align__(16) v2f wfrag[NSTEP][32];
    const int tid  = threadIdx.x;
    const int wave = tid >> 5;
    const int lane = tid & 31;
    fill_wfrag(wfrag, g1, tid);
    __syncthreads();

    const int gw    = blockIdx.x * WAVES_PER_BLOCK + wave;
    const int plane = gw >> 10;
    const int ty    = (gw >> 5) & 31;
    const int tx    = gw & 31;
    const int y0    = ty * 16, x0 = tx * 16;

    const float* pl = in + plane * PLANE;
    const int n  = lane & 15;
    const int hi = lane >> 4;
    const int kk = hi * 2;

    v8f acc = {0.f, 0.f, 0.f, 0.f, 0.f, 0.f, 0.f, 0.f};
    if (ty >= 2 && ty <= 29) {                 // interior: rows never reflect
        const float* src = pl + (y0 - RAD + kk) * IMG_W + x0 + n;
        #pragma unroll
        for (int s = 0; s < NSTEP; ++s) {
            v2f a = wfrag[s][lane];                        // Toeplitz weights
            v2f b;                                         // image rows (64B/row)
            b.x = src[(4 * s) * IMG_W];
            b.y = src[(4 * s + 1) * IMG_W];
            acc = __builtin_amdgcn_wmma_f32_16x16x4_f32(
                false, a, false, b, (short)0, acc, false, false);
        }
    } else {
        #pragma unroll
        for (int s = 0; s < NSTEP; ++s) {
            v2f a = wfrag[s][lane];
            const int r0 = reflect511(y0 - RAD + 4 * s + kk);
            const int r1 = reflect511(y0 - RAD + 4 * s + kk + 1);
            v2f b;
            b.x = pl[r0 * IMG_W + x0 + n];
            b.y = pl[r1 * IMG_W + x0 + n];
            acc = __builtin_amdgcn_wmma_f32_16x16x4_f32(
                false, a, false, b, (short)0, acc, false, false);
        }
    }

    const float* Pp = Pimg + plane * PLANE;
    if (MODE == 0) {
        float* Rp = Rbuf + plane * PLANE;
        float* Mp = outBuf + plane * PLANE;   // mask planar scratch lives in d_out
        #pragma unroll
        for (int r = 0; r < 8; ++r) {
            const int y = y0 + r + 8 * hi, x = x0 + n;
            const int idx = y * IMG_W + x;
            const float iv  = Pp[idx];
            const float res = iv - acc[r];
            Rp[idx] = res;
            Mp[idx] = (fabsf(res) * 255.f > 10.f) ? 1.f : 0.f;
        }
    } else {
        const float* Rp = Rbuf + plane * PLANE;
        const int b = plane / 3, c = plane - b * 3;
        #pragma unroll
        for (int r = 0; r < 8; ++r) {
            const int y = y0 + r + 8 * hi, x = x0 + n;
            const int idx = y * IMG_W + x;
            const float iv  = Pp[idx];
            const float res = Rp[idx];
            const float sm  = acc[r];
            const float sharp = fminf(fmaxf(iv + 0.5f * res, 0.f), 1.f);
            outBuf[((b * IMG_W + y) * IMG_W + x) * 3 + c] = sm * sharp + (1.f - sm) * iv;
        }
    }
}

// ---------------------------------------------------------------------------
extern "C" void kernel_launch(void* const* d_in, const int* in_sizes, int n_in,
                              void* d_out, int out_size, void* d_ws, size_t ws_size,
                              hipStream_t stream) {
    (void)in_sizes; (void)n_in; (void)out_size; (void)ws_size;
    const float* img = (const float*)d_in[0];   // (16,512,512,3) f32
    const float* k2d = (const float*)d_in[1];   // (51,51) f32
    float* out = (float*)d_out;

    // ws layout: g1[64] | P (img planar) | T (temp) | R (residual)  ~151 MB
    float* g1 = (float*)d_ws;
    float* P  = g1 + 64;
    float* T  = P + (size_t)NPLANES * PLANE;
    float* R  = T + (size_t)NPLANES * PLANE;

    const int elems  = NPLANES * PLANE;                    // 12.58M
    const int tileBlocks = TILES_TOTAL / WAVES_PER_BLOCK;  // 6144

    usm_gauss1d<<<1, 64, 0, stream>>>(k2d, g1);
    usm_nhwc2planar<<<elems / 256, 256, 0, stream>>>(img, P);

    usm_hblur<<<tileBlocks, 256, 0, stream>>>(P, T, g1);       // h-blur img
    usm_vblur<0><<<tileBlocks, 256, 0, stream>>>(T, P, g1, R, out); // -> res, mask
    usm_hblur<<<tileBlocks, 256, 0, stream>>>(out, T, g1);     // h-blur mask
    usm_vblur<1><<<tileBlocks, 256, 0, stream>>>(T, P, g1, R, out); // soft_mask+blend
}